// SAGPooling_39247411150919
// MI455X (gfx1250) — compile-verified
//
#include <hip/hip_runtime.h>
#include <stdint.h>

// ---------------------------------------------------------------------------
// Reference collapses algebraically:
//   P = one_hot(top_k(scores, N))  is a full permutation matrix (k == N)
//   adj_out = P^T (P adj) = adj           (bit-exact)
//   x_out   = P (P^T x)  = x  ->  x_out + x = 2*x   (bit-exact)
// Optimal kernel = pure streaming transform (536 MB total -> ~23 us @ 23.3
// TB/s). Both kernels use a branch-free full-tile fast path so 4 memory ops
// are in flight per wave; the adj stream uses the CDNA5 async global<->LDS
// path (ASYNCcnt) with 4-stage LDS buffering.
// ---------------------------------------------------------------------------

typedef __attribute__((ext_vector_type(4))) float v4f;

// Kernel 1: out = 2*x. Fast path: 4 independent 128-bit NT loads clustered,
// then double + NT store.
__global__ __launch_bounds__(256) void sag_scale2_f4(
    const v4f* __restrict__ x, v4f* __restrict__ out, uint32_t n4) {
  const uint32_t S = gridDim.x * blockDim.x;
  const uint32_t t = blockIdx.x * blockDim.x + threadIdx.x;
  const uint32_t i0 = t, i1 = t + S, i2 = t + 2u * S, i3 = t + 3u * S;

  if (i3 < n4) {  // all four chunks in-bounds (always true with our grid)
    v4f a = __builtin_nontemporal_load(&x[i0]);
    v4f b = __builtin_nontemporal_load(&x[i1]);
    v4f c = __builtin_nontemporal_load(&x[i2]);
    v4f d = __builtin_nontemporal_load(&x[i3]);
    __builtin_nontemporal_store(a + a, &out[i0]);
    __builtin_nontemporal_store(b + b, &out[i1]);
    __builtin_nontemporal_store(c + c, &out[i2]);
    __builtin_nontemporal_store(d + d, &out[i3]);
  } else {  // remainder (never taken with exact sizing; kept for safety)
    for (uint32_t i = i0; i < n4; i += S) {
      v4f v = __builtin_nontemporal_load(&x[i]);
      __builtin_nontemporal_store(v + v, &out[i]);
    }
  }
}

// Kernel 2: out = adj via the CDNA5 async data path. Four async loads into
// four disjoint LDS stage buffers (ASYNCcnt rises to 4), one
// s_wait_asynccnt 0, then four async stores. Each stage buffer is written
// once and read once per thread -> no LDS WAR hazards; trailing stores are
// drained by S_ENDPGM's implicit wait-idle.
__global__ __launch_bounds__(256) void sag_copy_adj_async(
    const v4f* __restrict__ src, v4f* __restrict__ dst, uint32_t n4) {
  __shared__ __align__(16) unsigned char lbuf[4][256 * 16];

  const uint32_t tid = threadIdx.x;
  const uint32_t S = gridDim.x * blockDim.x;
  const uint32_t t = blockIdx.x * blockDim.x + tid;
  const uint32_t i0 = t, i1 = t + S, i2 = t + 2u * S, i3 = t + 3u * S;

  // Flat LDS addresses are {SHARED_BASE[63:32], offset[31:0]} (ISA aperture
  // rules): truncating the generic pointer yields the LDS byte offset the
  // async ops take in their VDST/VSRC VGPR.
  const uint32_t l0 = (uint32_t)(uintptr_t)(void*)&lbuf[0][tid * 16];
  const uint32_t l1 = (uint32_t)(uintptr_t)(void*)&lbuf[1][tid * 16];
  const uint32_t l2 = (uint32_t)(uintptr_t)(void*)&lbuf[2][tid * 16];
  const uint32_t l3 = (uint32_t)(uintptr_t)(void*)&lbuf[3][tid * 16];

#define SAG_ALOAD(LO, GI)                                                  \
  {                                                                        \
    const void* ga_ = (const void*)(src + (GI));                           \
    asm volatile("global_load_async_to_lds_b128 %0, %1, off th:TH_LOAD_NT" \
                 :                                                         \
                 : "v"(LO), "v"(ga_)                                       \
                 : "memory");                                              \
  }
#define SAG_ASTORE(GI, LO)                                                 \
  {                                                                        \
    void* gd_ = (void*)(dst + (GI));                                       \
    asm volatile(                                                          \
        "global_store_async_from_lds_b128 %0, %1, off th:TH_STORE_NT"      \
        :                                                                  \
        : "v"(gd_), "v"(LO)                                                \
        : "memory");                                                       \
  }
#define SAG_AWAIT0() asm volatile("s_wait_asynccnt 0" ::: "memory")

  if (i3 < n4) {  // all four chunks in-bounds (always true with our grid)
    SAG_ALOAD(l0, i0)
    SAG_ALOAD(l1, i1)
    SAG_ALOAD(l2, i2)
    SAG_ALOAD(l3, i3)
    SAG_AWAIT0();  // all four loads complete before any LDS consumption
    SAG_ASTORE(i0, l0)
    SAG_ASTORE(i1, l1)
    SAG_ASTORE(i2, l2)
    SAG_ASTORE(i3, l3)
  } else {  // remainder (never taken with exact sizing; kept for safety)
    for (uint32_t i = i0; i < n4; i += S) {
      SAG_ALOAD(l0, i)
      SAG_AWAIT0();
      SAG_ASTORE(i, l0)
      SAG_AWAIT0();  // store done before l0 is overwritten next iteration
    }
  }

#undef SAG_ALOAD
#undef SAG_ASTORE
#undef SAG_AWAIT0
}

extern "C" void kernel_launch(void* const* d_in, const int* in_sizes, int n_in,
                              void* d_out, int out_size, void* d_ws, size_t ws_size,
                              hipStream_t stream) {
  const float* x = (const float*)d_in[0];    // [B, N, N] fp32
  const float* adj = (const float*)d_in[1];  // [B, N, N] fp32
  // d_in[2] (W) and d_in[3] (b) are provably unused: the score/sort/one-hot
  // pipeline composes to the identity permutation product P P^T = I.
  float* out = (float*)d_out;                // [2 * B*N*N] fp32

  const uint32_t n = (uint32_t)in_sizes[0];  // B*N*N = 33,554,432 (mult. of 4)
  const uint32_t n4 = n / 4u;

  const uint32_t block = 256u;
  const uint32_t chunks_per_thread = 4u;
  const uint32_t grid =
      (n4 + block * chunks_per_thread - 1u) / (block * chunks_per_thread);
  // n4 = 8,388,608 -> grid = 8192 blocks, every chunk in-bounds.

  sag_scale2_f4<<<dim3(grid), dim3(block), 0, stream>>>(
      (const v4f*)x, (v4f*)out, n4);
  sag_copy_adj_async<<<dim3(grid), dim3(block), 0, stream>>>(
      (const v4f*)adj, (v4f*)(out + n), n4);
}